// GraphAttentionLayer_61392262529420
// MI455X (gfx1250) — compile-verified
//
#include <hip/hip_runtime.h>
#include <hip/hip_bf16.h>
#include <math.h>

typedef __attribute__((ext_vector_type(2))) float v2f;
typedef __attribute__((ext_vector_type(8))) float v8f;

#define ALPHA 0.2f

// ---------------------------------------------------------------------------
// Kernel 0: initialize seg_max = -inf, denom = 0, out = 0
// ---------------------------------------------------------------------------
__global__ void gat_init(float* __restrict__ seg_max, float* __restrict__ denom,
                         float* __restrict__ out, int n_nodes) {
    int i = blockIdx.x * blockDim.x + threadIdx.x;
    if (i < n_nodes) {
        seg_max[i] = -INFINITY;
        denom[i]   = 0.0f;
    }
    if (i < n_nodes * 64) {
        out[i] = 0.0f;
    }
}

// ---------------------------------------------------------------------------
// Kernel 1: h = x @ W^T via V_WMMA_F32_16X16X4_F32.
// One wave computes a 16-row x 64-col tile of h, accumulating K=64 in 16
// steps of K=4, for 4 column tiles of N=16 -> 64 WMMAs per wave.
//
// A (16x4, f32):  lanes 0-15 hold M=lane, VGPR0/1 = K{0,1}; lanes 16-31 hold
//                 M=lane-16, VGPR0/1 = K{2,3}   (ISA 7.12.2)
// B (4x16, f32):  row K striped across lanes: VGPR0 = K0 (lanes 0-15) /
//                 K2 (lanes 16-31), VGPR1 = K1 / K3.  B[k][n] = W[n][k].
// C/D (16x16):    VGPR v: M = v + 8*half, N = lane&15.
// ---------------------------------------------------------------------------
__global__ void gat_wmma_gemm(const float* __restrict__ x,
                              const float* __restrict__ W,
                              float* __restrict__ h, int n_nodes) {
    const int wave = (blockIdx.x * blockDim.x + threadIdx.x) >> 5;
    const int lane = threadIdx.x & 31;
    const int m0   = wave * 16;
    if (m0 >= n_nodes) return;              // wave-uniform guard: EXEC all-1s inside

    const int half = lane >> 4;
    const int r    = lane & 15;

    v8f acc0 = {}, acc1 = {}, acc2 = {}, acc3 = {};

#pragma unroll
    for (int kk = 0; kk < 16; ++kk) {
        const int kb = kk * 4 + half * 2;   // this half-wave's K pair
        v2f a;
        a.x = x[(m0 + r) * 64 + kb + 0];
        a.y = x[(m0 + r) * 64 + kb + 1];

        v2f b0, b1, b2, b3;
        b0.x = W[(0 * 16 + r) * 64 + kb + 0];  b0.y = W[(0 * 16 + r) * 64 + kb + 1];
        b1.x = W[(1 * 16 + r) * 64 + kb + 0];  b1.y = W[(1 * 16 + r) * 64 + kb + 1];
        b2.x = W[(2 * 16 + r) * 64 + kb + 0];  b2.y = W[(2 * 16 + r) * 64 + kb + 1];
        b3.x = W[(3 * 16 + r) * 64 + kb + 0];  b3.y = W[(3 * 16 + r) * 64 + kb + 1];

        acc0 = __builtin_amdgcn_wmma_f32_16x16x4_f32(false, a, false, b0, (short)0, acc0, false, false);
        acc1 = __builtin_amdgcn_wmma_f32_16x16x4_f32(false, a, false, b1, (short)0, acc1, false, false);
        acc2 = __builtin_amdgcn_wmma_f32_16x16x4_f32(false, a, false, b2, (short)0, acc2, false, false);
        acc3 = __builtin_amdgcn_wmma_f32_16x16x4_f32(false, a, false, b3, (short)0, acc3, false, false);
    }

#pragma unroll
    for (int v = 0; v < 8; ++v) {
        const int row = (m0 + v + 8 * half) * 64;
        h[row +  0 + r] = acc0[v];
        h[row + 16 + r] = acc1[v];
        h[row + 32 + r] = acc2[v];
        h[row + 48 + r] = acc3[v];
    }
}

// ---------------------------------------------------------------------------
// Kernel 2: per-node scores  s1[n] = h[n] . a[0:64],  s2[n] = h[n] . a[64:128]
// One wave32 per node; lane l covers components l and l+32; shuffle-reduce.
// ---------------------------------------------------------------------------
__global__ void gat_scores(const float* __restrict__ h,
                           const float* __restrict__ a_vec,
                           float* __restrict__ s1, float* __restrict__ s2,
                           int n_nodes) {
    const int node = (blockIdx.x * blockDim.x + threadIdx.x) >> 5;
    const int lane = threadIdx.x & 31;
    if (node >= n_nodes) return;

    const float h0 = h[node * 64 + lane];
    const float h1 = h[node * 64 + 32 + lane];
    float p1 = h0 * a_vec[lane]      + h1 * a_vec[32 + lane];
    float p2 = h0 * a_vec[64 + lane] + h1 * a_vec[96 + lane];

#pragma unroll
    for (int off = 16; off > 0; off >>= 1) {
        p1 += __shfl_down(p1, off, 32);
        p2 += __shfl_down(p2, off, 32);
    }
    if (lane == 0) {
        s1[node] = p1;
        s2[node] = p2;
    }
}

__device__ __forceinline__ float leaky(float v) {
    return v > 0.0f ? v : ALPHA * v;
}

// ---------------------------------------------------------------------------
// Kernel 3: per-edge score -> segment max over src (float atomic max -> L2)
// ---------------------------------------------------------------------------
__global__ void gat_edge_max(const long long* __restrict__ srcI,
                             const long long* __restrict__ dstI,
                             const float* __restrict__ s1,
                             const float* __restrict__ s2,
                             float* __restrict__ seg_max, int n_edges) {
    const int e = blockIdx.x * blockDim.x + threadIdx.x;
    if (e >= n_edges) return;
    const int s = (int)srcI[e];
    const int d = (int)dstI[e];
    const float v = leaky(s1[s] + s2[d]);
    atomicMax(&seg_max[s], v);
}

// ---------------------------------------------------------------------------
// Kernel 4: exp_e = exp(e - seg_max[src]); denom[src] += exp_e
// ---------------------------------------------------------------------------
__global__ void gat_edge_exp(const long long* __restrict__ srcI,
                             const long long* __restrict__ dstI,
                             const float* __restrict__ s1,
                             const float* __restrict__ s2,
                             const float* __restrict__ seg_max,
                             float* __restrict__ exp_e,
                             float* __restrict__ denom, int n_edges) {
    const int e = blockIdx.x * blockDim.x + threadIdx.x;
    if (e >= n_edges) return;
    const int s = (int)srcI[e];
    const int d = (int)dstI[e];
    const float v  = leaky(s1[s] + s2[d]);
    const float ex = expf(v - seg_max[s]);
    exp_e[e] = ex;
    atomicAdd(&denom[s], ex);
}

// ---------------------------------------------------------------------------
// Kernel 5: scatter  out[dst] += (exp_e/denom[src]) * h[src]
// One wave32 per edge; lane l handles components l and l+32 (coalesced 128B
// gathers of h and coalesced f32 atomic adds resolved in L2).
// ---------------------------------------------------------------------------
__global__ void gat_scatter(const long long* __restrict__ srcI,
                            const long long* __restrict__ dstI,
                            const float* __restrict__ h,
                            const float* __restrict__ exp_e,
                            const float* __restrict__ denom,
                            float* __restrict__ out, int n_edges) {
    const int e    = (blockIdx.x * blockDim.x + threadIdx.x) >> 5;
    const int lane = threadIdx.x & 31;
    if (e >= n_edges) return;
    const int s = (int)srcI[e];
    const int d = (int)dstI[e];
    const float att = exp_e[e] / denom[s];
    const float v0 = att * h[s * 64 + lane];
    const float v1 = att * h[s * 64 + 32 + lane];
    atomicAdd(&out[d * 64 + lane],      v0);
    atomicAdd(&out[d * 64 + 32 + lane], v1);
}

// ---------------------------------------------------------------------------
// Kernel 6: in-place ELU
// ---------------------------------------------------------------------------
__global__ void gat_elu(float* __restrict__ out, int n) {
    const int i = blockIdx.x * blockDim.x + threadIdx.x;
    if (i >= n) return;
    const float v = out[i];
    out[i] = v > 0.0f ? v : (expf(v) - 1.0f);
}

extern "C" void kernel_launch(void* const* d_in, const int* in_sizes, int n_in,
                              void* d_out, int out_size, void* d_ws, size_t ws_size,
                              hipStream_t stream) {
    const float*     x    = (const float*)d_in[0];
    const long long* eidx = (const long long*)d_in[1];   // int64 (2, E)
    const float*     W    = (const float*)d_in[2];
    const float*     a    = (const float*)d_in[3];
    float*           out  = (float*)d_out;

    const int n_nodes = in_sizes[0] / 64;
    const int n_edges = in_sizes[1] / 2;
    const long long* srcI = eidx;            // row 0
    const long long* dstI = eidx + n_edges;  // row 1

    // workspace layout (floats)
    float* h       = (float*)d_ws;                 // n_nodes*64
    float* s1      = h + (size_t)n_nodes * 64;     // n_nodes
    float* s2      = s1 + n_nodes;                 // n_nodes
    float* seg_max = s2 + n_nodes;                 // n_nodes
    float* denom   = seg_max + n_nodes;            // n_nodes
    float* exp_e   = denom + n_nodes;              // n_edges

    const int BT = 256;

    // 0: init accumulators
    {
        int n = n_nodes * 64;
        gat_init<<<(n + BT - 1) / BT, BT, 0, stream>>>(seg_max, denom, out, n_nodes);
    }
    // 1: h = x @ W^T  (WMMA, one wave per 16 rows)
    {
        int waves  = (n_nodes + 15) / 16;
        int blocks = (waves * 32 + BT - 1) / BT;
        gat_wmma_gemm<<<blocks, BT, 0, stream>>>(x, W, h, n_nodes);
    }
    // 2: per-node attention scores (one wave per node)
    {
        int blocks = ((n_nodes * 32) + BT - 1) / BT;
        gat_scores<<<blocks, BT, 0, stream>>>(h, a, s1, s2, n_nodes);
    }
    // 3: segment max over src
    gat_edge_max<<<(n_edges + BT - 1) / BT, BT, 0, stream>>>(srcI, dstI, s1, s2, seg_max, n_edges);
    // 4: exp + denom
    gat_edge_exp<<<(n_edges + BT - 1) / BT, BT, 0, stream>>>(srcI, dstI, s1, s2, seg_max, exp_e, denom, n_edges);
    // 5: weighted scatter (one wave per edge)
    {
        long long threads = (long long)n_edges * 32;
        int blocks = (int)((threads + BT - 1) / BT);
        gat_scatter<<<blocks, BT, 0, stream>>>(srcI, dstI, h, exp_e, denom, out, n_edges);
    }
    // 6: ELU
    {
        int n = n_nodes * 64;
        gat_elu<<<(n + BT - 1) / BT, BT, 0, stream>>>(out, n);
    }
}